// SoftToHardNdEncoder_27608049779090
// MI455X (gfx1250) — compile-verified
//
#include <hip/hip_runtime.h>

// Soft-to-hard VQ encoder for MI455X (gfx1250, wave32).
// GEMM1 (distances) and GEMM2 (soft symbols) via V_WMMA_F32_16X16X4_F32.

typedef float v2f __attribute__((ext_vector_type(2)));
typedef float v8f __attribute__((ext_vector_type(8)));

#define B_    4
#define C_    128
#define H_    64
#define W_    64
#define L_    16
#define K_    64
#define CD_   8
#define HW_   (H_ * W_)        // 4096
#define NPIX  (B_ * HW_)       // 16384
#define QSIZE (B_ * C_ * HW_)  // 2097152 floats (quantized), idxes follow
#define WAVES 8
#define THREADS (WAVES * 32)

__global__ __launch_bounds__(THREADS)
void vq_wmma_kernel(const float* __restrict__ z,
                    const float* __restrict__ codes,
                    float* __restrict__ out) {
  extern __shared__ float smem[];
  float* s_codesT = smem;                 // [L][16][64] codes transposed, d=8..15 zero  (64 KB)
  float* s_cnorm  = smem + L_ * 16 * 64;  // [L][64] per-code |c|^2                      (4 KB)
  float* s_P      = s_cnorm + L_ * 64;    // [WAVES][16][64] softmax probs staging       (32 KB)

  const int tid = threadIdx.x;

  // ---- preload codes into LDS: transpose [l][k][d] -> [l][d][64], plus |c|^2 ----
  for (int i = tid; i < L_ * K_; i += THREADS) {
    const int l = i >> 6, k = i & 63;
    float nrm = 0.f;
#pragma unroll
    for (int d = 0; d < CD_; ++d) {
      const float c = codes[i * CD_ + d];
      s_codesT[(l * 16 + d) * 64 + k] = c;
      nrm += c * c;
    }
    s_cnorm[i] = nrm;
  }
  // zero-pad the d = 8..15 stripes (lets GEMM2's 16-wide B tile be unconditional)
  for (int j = tid; j < L_ * 8 * 64; j += THREADS) {
    const int l = j >> 9, off = j & 511;
    s_codesT[l * 1024 + 512 + off] = 0.f;
  }
  __syncthreads();

  const int wave = tid >> 5;
  const int lane = tid & 31;
  const int n    = lane & 15;   // column index (code k % 16 / channel d)
  const int hi   = lane >> 4;   // which 16-lane half

  const int task = blockIdx.x * WAVES + wave;  // 0..16383
  const int tile = task >> 4;                  // 16-pixel tile
  const int l    = task & 15;                  // latent slot
  const int p0   = tile << 4;
  const int b    = p0 >> 12;                   // 4096 pixels per batch image
  const int hw0  = p0 & (HW_ - 1);

  // ---- A operand: X (16 pixels x 8 dims), f32 A-layout:
  //      lane m (lanes 0-15): K = 0,1 ; lanes 16-31: K = 2,3 per 16x16x4 step
  const int zbase = b * C_ * HW_ + hw0 + n;    // pixel row m = n
  v2f A0, A1;
  A0.x = z[zbase + (l * CD_ + 2 * hi    ) * HW_];
  A0.y = z[zbase + (l * CD_ + 2 * hi + 1) * HW_];
  A1.x = z[zbase + (l * CD_ + 2 * hi + 4) * HW_];
  A1.y = z[zbase + (l * CD_ + 2 * hi + 5) * HW_];
  const float part  = A0.x * A0.x + A0.y * A0.y + A1.x * A1.x + A1.y * A1.y;
  const float xnorm = part + __shfl_xor(part, 16, 32);  // full |x_m|^2 at lanes m and m+16

  // ---- GEMM1: S = X * C^T  (16x64), 4 N-tiles x 2 K-chunks = 8 WMMAs ----
  v8f   acc[4];
  float cn[4];
#pragma unroll
  for (int t = 0; t < 4; ++t) {
    const int k = 16 * t + n;  // this lane's code column
    v2f b0, b1;                // B layout: lane = N, low/high half = K 0..1 / 2..3
    b0.x = s_codesT[(l * 16 + 2 * hi    ) * 64 + k];
    b0.y = s_codesT[(l * 16 + 2 * hi + 1) * 64 + k];
    b1.x = s_codesT[(l * 16 + 2 * hi + 4) * 64 + k];
    b1.y = s_codesT[(l * 16 + 2 * hi + 5) * 64 + k];
    v8f zeroc = {};
    acc[t] = __builtin_amdgcn_wmma_f32_16x16x4_f32(false, A0, false, b0,
                                                   (short)0, zeroc, false, false);
    acc[t] = __builtin_amdgcn_wmma_f32_16x16x4_f32(false, A1, false, b1,
                                                   (short)0, acc[t], false, false);
    cn[t] = s_cnorm[l * 64 + k];
  }

  // ---- distances, argmin, softmax (row m = r + 8*hi lives in half 'hi') ----
  int kmin[8];
#pragma unroll
  for (int r = 0; r < 8; ++r) {
    const float xn = __shfl(xnorm, r + 8 * hi, 32);
    float dd[4];
#pragma unroll
    for (int t = 0; t < 4; ++t)
      dd[t] = sqrtf(fmaxf(xn + cn[t] - 2.f * acc[t][r], 0.f));

    float mn = dd[0];
    int   mi = n;
#pragma unroll
    for (int t = 1; t < 4; ++t)
      if (dd[t] < mn) { mn = dd[t]; mi = 16 * t + n; }
    // butterfly over the 16 lanes of this half; ties -> lowest index (matches argmin)
#pragma unroll
    for (int mask = 1; mask <= 8; mask <<= 1) {
      const float om = __shfl_xor(mn, mask, 32);
      const int   oi = __shfl_xor(mi, mask, 32);
      if (om < mn || (om == mn && oi < mi)) { mn = om; mi = oi; }
    }

    float e[4], es = 0.f;
#pragma unroll
    for (int t = 0; t < 4; ++t) { e[t] = __expf(mn - dd[t]); es += e[t]; }
#pragma unroll
    for (int mask = 1; mask <= 8; mask <<= 1) es += __shfl_xor(es, mask, 32);
    const float inv = 1.f / es;

    const int m = r + 8 * hi;
#pragma unroll
    for (int t = 0; t < 4; ++t)
      s_P[(wave * 16 + m) * 64 + 16 * t + n] = e[t] * inv;  // row-major P for GEMM2 A
    kmin[r] = mi;
  }
  // same-wave DS ops are in order; stores above complete before the loads below

  // ---- GEMM2: soft = P(16x64) * C(64x8 padded to 16), 16 K-steps ----
  v8f acc2 = {};
#pragma unroll
  for (int s = 0; s < 16; ++s) {
    const v2f pa = *(const v2f*)&s_P[(wave * 16 + n) * 64 + 4 * s + 2 * hi];
    const v2f pb = *(const v2f*)&s_codesT[(l * 16 + n) * 64 + 4 * s + 2 * hi];
    acc2 = __builtin_amdgcn_wmma_f32_16x16x4_f32(false, pa, false, pb,
                                                 (short)0, acc2, false, false);
  }

  // ---- outputs: quantized = (hard - soft) + soft, plus idxes as float ----
  const int d = n;  // output column N = channel dim
#pragma unroll
  for (int r = 0; r < 8; ++r) {
    const int m   = r + 8 * hi;
    const int pix = p0 + m;
    if (d < CD_) {
      const float hard = s_codesT[(l * 16 + d) * 64 + kmin[r]];
      const float soft = acc2[r];
      out[b * C_ * HW_ + (l * CD_ + d) * HW_ + hw0 + m] = (hard - soft) + soft;
    }
    if (n == 0)  // one writer per row-half
      out[QSIZE + pix * L_ + l] = (float)kmin[r];
  }
}

extern "C" void kernel_launch(void* const* d_in, const int* in_sizes, int n_in,
                              void* d_out, int out_size, void* d_ws, size_t ws_size,
                              hipStream_t stream) {
  const float* z     = (const float*)d_in[0];
  const float* codes = (const float*)d_in[1];
  float*       out   = (float*)d_out;

  const int    ntasks = (NPIX / 16) * L_;          // 16384 wave-tasks
  const int    blocks = ntasks / WAVES;            // 2048 blocks of 8 waves
  const size_t shmem  = (size_t)(L_ * 16 * 64 + L_ * 64 + WAVES * 16 * 64) * sizeof(float); // 100 KB

  vq_wmma_kernel<<<blocks, THREADS, shmem, stream>>>(z, codes, out);
}